// TensorNet2_61924838474267
// MI455X (gfx1250) — compile-verified
//
#include <hip/hip_runtime.h>
#include <math.h>

// ---------------- problem constants ----------------
#define N_NODES 10000
#define N_EDGES 160000
#define H       128
#define Q       16
#define RBF     32
#define DIN     (RBF + 2*Q)        // 64
#define NC      (N_NODES * H)      // one component plane: 1.28M floats
#define CUTOFF  4.5f
#define ET      64                 // edges per block in the fused MLP kernel
#define BM      128                // rows per block in k_mix
#define N_MBLK  ((N_NODES + BM - 1) / BM)   // 79 (last block partial)

typedef float v2f __attribute__((ext_vector_type(2)));
typedef float v8f __attribute__((ext_vector_type(8)));

// ---------------- fp32 WMMA helpers (V_WMMA_F32_16X16X4_F32) ----------------
__device__ __forceinline__ v8f wmma4(v2f a, v2f b, v8f c) {
    // 8 args: (neg_a, A, neg_b, B, c_mod, C, reuse_a, reuse_b)
    return __builtin_amdgcn_wmma_f32_16x16x4_f32(false, a, false, b, (short)0, c, false, false);
}

// A fragment (16x4 f32): lanes 0-15 hold M=0..15 K={k0,k0+1}; lanes 16-31 K={k0+2,k0+3}
__device__ __forceinline__ v2f frag_a(const float* base, int ld, int m0, int k0, int lane) {
    int m = m0 + (lane & 15);
    int k = k0 + ((lane >> 4) << 1);
    v2f a; a.x = base[m * ld + k]; a.y = base[m * ld + k + 1]; return a;
}
// B fragment (4x16 f32), row-major W[k][n]: lanes hold N, VGPR pair holds K halves
__device__ __forceinline__ v2f frag_b(const float* W, int ldn, int n0, int k0, int lane) {
    int n = n0 + (lane & 15);
    int k = k0 + ((lane >> 4) << 1);
    v2f b; b.x = W[k * ldn + n]; b.y = W[(k + 1) * ldn + n]; return b;
}

__device__ __forceinline__ float silu(float v) { return v * (1.0f / (1.0f + __expf(-v))); }

// ---------------- gfx1250 async global->LDS copies (ASYNCcnt-tracked) ----------------
// LDS byte offset = low 32 bits of the generic address (ISA aperture: {SHARED_BASE, off[31:0]})
__device__ __forceinline__ void async_ld_b32(void* lds_ptr, const void* gptr) {
    unsigned loff = (unsigned)(uintptr_t)lds_ptr;
    unsigned long long ga = (unsigned long long)(uintptr_t)gptr;
    asm volatile("global_load_async_to_lds_b32 %0, %1, off" :: "v"(loff), "v"(ga) : "memory");
}
__device__ __forceinline__ void async_ld_b128(void* lds_ptr, const void* gptr) {
    unsigned loff = (unsigned)(uintptr_t)lds_ptr;
    unsigned long long ga = (unsigned long long)(uintptr_t)gptr;
    asm volatile("global_load_async_to_lds_b128 %0, %1, off" :: "v"(loff), "v"(ga) : "memory");
}
__device__ __forceinline__ void wait_async0() {
    asm volatile("s_wait_asynccnt 0" ::: "memory");
}

// ---------------- kernel: normalize + decompose X ----------------
// planes layout (component-major): [0]=I, [1..3]=skew vec, [4..9]=sym (00,01,02,11,12,22)
__global__ void k_decompose(const float* __restrict__ X, float* __restrict__ Xn,
                            float* __restrict__ plane) {
    int idx = blockIdx.x * blockDim.x + threadIdx.x;   // node*H + channel
    if (idx >= NC) return;
    const float* xp = X + (size_t)idx * 9;
    float m[9]; float nrm = 0.0f;
#pragma unroll
    for (int i = 0; i < 9; ++i) { m[i] = xp[i]; nrm += m[i] * m[i]; }
    float inv = 1.0f / (nrm + 1.0f);
#pragma unroll
    for (int i = 0; i < 9; ++i) { m[i] *= inv; Xn[(size_t)idx * 9 + i] = m[i]; }
    float I = (m[0] + m[4] + m[8]) * (1.0f / 3.0f);
    plane[0 * NC + idx] = I;
    plane[1 * NC + idx] = 0.5f * (m[7] - m[5]);
    plane[2 * NC + idx] = 0.5f * (m[2] - m[6]);
    plane[3 * NC + idx] = 0.5f * (m[3] - m[1]);
    plane[4 * NC + idx] = m[0] - I;
    plane[5 * NC + idx] = 0.5f * (m[1] + m[3]);
    plane[6 * NC + idx] = 0.5f * (m[2] + m[6]);
    plane[7 * NC + idx] = m[4] - I;
    plane[8 * NC + idx] = 0.5f * (m[5] + m[7]);
    plane[9 * NC + idx] = m[8] - I;
}

// ---------------- kernel: per-component channel mixing GEMM (WMMA f32) ----------------
// out[c][n,g] = sum_h in[c][n,h] * Wt[sel(c)][h,g]
// Block: 128 rows x 128 cols; A tile staged in LDS via async b128; each wave owns one
// 16-col N-tile and 8 M-tile accumulators so each B fragment is loaded once per k-step.
__global__ void k_mix(const float* __restrict__ in_plane, const float* __restrict__ Wt,
                      float* __restrict__ out_plane, int w0, int w1, int w2) {
    __shared__ float A_s[BM * H];          // 64 KB
    int comp = blockIdx.y;
    int tid  = threadIdx.x;
    int lane = tid & 31;
    int wave = tid >> 5;                   // 8 waves -> 8 N-tiles of 16
    int rowbase = blockIdx.x * BM;
    const float* A = in_plane + (size_t)comp * NC;
    int wi = (comp == 0) ? w0 : (comp < 4 ? w1 : w2);
    const float* W = Wt + (size_t)wi * H * H;

    // stage A tile: 128x128 floats as 4096 b128 async copies (clamped rows for tail block)
    for (int f = tid; f < (BM * H) / 4; f += 256) {
        int row = f >> 5;                  // 32 float4 per row
        int c4  = f & 31;
        int gr  = rowbase + row;
        if (gr >= N_NODES) gr = N_NODES - 1;
        async_ld_b128(&A_s[row * H + c4 * 4], A + (size_t)gr * H + c4 * 4);
    }
    wait_async0();
    __syncthreads();

    int n0 = wave * 16;
    v8f acc[8];
    { v8f z = {}; for (int mt = 0; mt < 8; ++mt) acc[mt] = z; }
    for (int k0 = 0; k0 < H; k0 += 4) {
        v2f b = frag_b(W, H, n0, k0, lane);
#pragma unroll
        for (int mt = 0; mt < 8; ++mt)
            acc[mt] = wmma4(frag_a(A_s, H, mt * 16, k0, lane), b, acc[mt]);
    }

    float* D = out_plane + (size_t)comp * NC;
    int half = lane >> 4, col = n0 + (lane & 15);
    for (int mt = 0; mt < 8; ++mt) {
#pragma unroll
        for (int r = 0; r < 8; ++r) {
            int gr = rowbase + mt * 16 + r + 8 * half;
            if (gr < N_NODES) D[(size_t)gr * H + col] = acc[mt][r];
        }
    }
}

// ---------------- kernel: zero message accumulators ----------------
__global__ void k_zero(float* __restrict__ p, int n) {
    int i = blockIdx.x * blockDim.x + threadIdx.x;
    if (i < n) p[i] = 0.0f;
}

// ---------------- fused kernel: edge MLP (3x WMMA GEMM) + gather/scatter message ----------------
__global__ void k_edge(const float* __restrict__ charges, const float* __restrict__ ew,
                       const float* __restrict__ ea,
                       const float* __restrict__ W1, const float* __restrict__ b1,
                       const float* __restrict__ W2, const float* __restrict__ b2,
                       const float* __restrict__ W3, const float* __restrict__ b3,
                       const int* __restrict__ eidx,
                       const float* __restrict__ mixed, float* __restrict__ msg) {
    __shared__ float in_s[ET * DIN];       // 16 KB
    __shared__ float h1_s[ET * H];         // 32 KB
    __shared__ float h2_s[ET * 2 * H];     // 64 KB
    __shared__ float x_s[ET * 3 * H];      // 96 KB
    __shared__ int   src_s[ET], dst_s[ET];
    __shared__ float c_s[ET];

    int tid = threadIdx.x;
    int lane = tid & 31, wave = tid >> 5;
    int e0 = blockIdx.x * ET;

    if (tid < ET) {
        int e = e0 + tid;
        src_s[tid] = eidx[e];
        dst_s[tid] = eidx[N_EDGES + e];
        float w = ew[e];
        float c = 0.5f * (__cosf(w * (3.14159265358979f / CUTOFF)) + 1.0f);
        c_s[tid] = (w < CUTOFF) ? c : 0.0f;
    }
    __syncthreads();

    // stage input tile [ET, 64] = [edge_attr | charges[src] | charges[dst]] via async b32
    for (int i = tid; i < ET * DIN; i += 256) {
        int e = i >> 6, c = i & 63;
        const float* g;
        if (c < RBF)          g = ea + (size_t)(e0 + e) * RBF + c;
        else if (c < RBF + Q) g = charges + src_s[e] * Q + (c - RBF);
        else                  g = charges + dst_s[e] * Q + (c - RBF - Q);
        async_ld_b32(&in_s[i], g);
    }
    wait_async0();
    __syncthreads();

    // ---- layer 1: [ET,64] @ [64,128] -> h1, silu.  1 N-tile per wave, 4 M-tiles ----
    {
        int n0 = wave * 16;
        v8f acc[4];
        { v8f z = {}; for (int mt = 0; mt < 4; ++mt) acc[mt] = z; }
        for (int k0 = 0; k0 < DIN; k0 += 4) {
            v2f b = frag_b(W1, H, n0, k0, lane);
#pragma unroll
            for (int mt = 0; mt < 4; ++mt)
                acc[mt] = wmma4(frag_a(in_s, DIN, mt * 16, k0, lane), b, acc[mt]);
        }
        int half = lane >> 4, col = n0 + (lane & 15);
        float bias = b1[col];
        for (int mt = 0; mt < 4; ++mt)
#pragma unroll
            for (int r = 0; r < 8; ++r)
                h1_s[(mt * 16 + r + 8 * half) * H + col] = silu(acc[mt][r] + bias);
    }
    __syncthreads();

    // ---- layer 2: [ET,128] @ [128,256] -> h2, silu.  2 N-tiles per wave ----
    for (int nt = wave; nt < 16; nt += 8) {
        int n0 = nt * 16;
        v8f acc[4];
        { v8f z = {}; for (int mt = 0; mt < 4; ++mt) acc[mt] = z; }
        for (int k0 = 0; k0 < H; k0 += 4) {
            v2f b = frag_b(W2, 2 * H, n0, k0, lane);
#pragma unroll
            for (int mt = 0; mt < 4; ++mt)
                acc[mt] = wmma4(frag_a(h1_s, H, mt * 16, k0, lane), b, acc[mt]);
        }
        int half = lane >> 4, col = n0 + (lane & 15);
        float bias = b2[col];
        for (int mt = 0; mt < 4; ++mt)
#pragma unroll
            for (int r = 0; r < 8; ++r)
                h2_s[(mt * 16 + r + 8 * half) * 2 * H + col] = silu(acc[mt][r] + bias);
    }
    __syncthreads();

    // ---- layer 3: [ET,256] @ [256,384] -> x, silu, * cutoff.  3 N-tiles per wave ----
    for (int nt = wave; nt < 24; nt += 8) {
        int n0 = nt * 16;
        v8f acc[4];
        { v8f z = {}; for (int mt = 0; mt < 4; ++mt) acc[mt] = z; }
        for (int k0 = 0; k0 < 2 * H; k0 += 4) {
            v2f b = frag_b(W3, 3 * H, n0, k0, lane);
#pragma unroll
            for (int mt = 0; mt < 4; ++mt)
                acc[mt] = wmma4(frag_a(h2_s, 2 * H, mt * 16, k0, lane), b, acc[mt]);
        }
        int half = lane >> 4, col = n0 + (lane & 15);
        float bias = b3[col];
        for (int mt = 0; mt < 4; ++mt)
#pragma unroll
            for (int r = 0; r < 8; ++r) {
                int row = mt * 16 + r + 8 * half;
                x_s[row * 3 * H + col] = silu(acc[mt][r] + bias) * c_s[row];
            }
    }
    __syncthreads();

    // ---- message: gather mixed[dst], scale by x, scatter-add into msg[src] ----
    int ch = tid & (H - 1);
    for (int e = tid >> 7; e < ET; e += 2) {
        int s = src_s[e], d = dst_s[e];
        size_t sb = (size_t)s * H + ch;
        size_t db = (size_t)d * H + ch;
        float x0 = x_s[e * 3 * H + ch];
        float x1 = x_s[e * 3 * H + H + ch];
        float x2 = x_s[e * 3 * H + 2 * H + ch];
        atomicAdd(&msg[0 * NC + sb], x0 * mixed[0 * NC + db]);
#pragma unroll
        for (int k = 0; k < 3; ++k)
            atomicAdd(&msg[(1 + k) * NC + sb], x1 * mixed[(1 + k) * NC + db]);
#pragma unroll
        for (int k = 0; k < 6; ++k)
            atomicAdd(&msg[(4 + k) * NC + sb], x2 * mixed[(4 + k) * NC + db]);
    }
}

// ---------------- kernel: O(3) recombination + decompose + norm-scale ----------------
__device__ __forceinline__ void build33(const float* __restrict__ p, int idx, float* M) {
    float I  = p[0 * NC + idx];
    float a0 = p[1 * NC + idx], a1 = p[2 * NC + idx], a2 = p[3 * NC + idx];
    float s00 = p[4 * NC + idx], s01 = p[5 * NC + idx], s02 = p[6 * NC + idx];
    float s11 = p[7 * NC + idx], s12 = p[8 * NC + idx], s22 = p[9 * NC + idx];
    M[0] = I + s00;  M[1] = s01 - a2; M[2] = s02 + a1;
    M[3] = s01 + a2; M[4] = I + s11;  M[5] = s12 - a0;
    M[6] = s02 - a1; M[7] = s12 + a0; M[8] = I + s22;
}

__global__ void k_recombine(const float* __restrict__ msg, const float* __restrict__ mixed,
                            float* __restrict__ scaled) {
    int idx = blockIdx.x * blockDim.x + threadIdx.x;
    if (idx >= NC) return;
    float M[9], Y[9], T[9];
    build33(msg, idx, M);
    build33(mixed, idx, Y);
#pragma unroll
    for (int i = 0; i < 3; ++i)
#pragma unroll
        for (int j = 0; j < 3; ++j) {
            float t = 0.0f;
#pragma unroll
            for (int k = 0; k < 3; ++k)
                t += M[i * 3 + k] * Y[k * 3 + j] + Y[i * 3 + k] * M[k * 3 + j];
            T[i * 3 + j] = t;
        }
    float nrm = 0.0f;
#pragma unroll
    for (int i = 0; i < 9; ++i) nrm += T[i] * T[i];   // _X == T exactly
    float inv = 1.0f / (nrm + 1.0f);
    float I2 = (T[0] + T[4] + T[8]) * (1.0f / 3.0f);
    scaled[0 * NC + idx] = I2 * inv;
    scaled[1 * NC + idx] = 0.5f * (T[7] - T[5]) * inv;
    scaled[2 * NC + idx] = 0.5f * (T[2] - T[6]) * inv;
    scaled[3 * NC + idx] = 0.5f * (T[3] - T[1]) * inv;
    scaled[4 * NC + idx] = (T[0] - I2) * inv;
    scaled[5 * NC + idx] = 0.5f * (T[1] + T[3]) * inv;
    scaled[6 * NC + idx] = 0.5f * (T[2] + T[6]) * inv;
    scaled[7 * NC + idx] = (T[4] - I2) * inv;
    scaled[8 * NC + idx] = 0.5f * (T[5] + T[7]) * inv;
    scaled[9 * NC + idx] = (T[8] - I2) * inv;
}

// ---------------- kernel: dX rebuild, out = Xn + dX + dX@dX ----------------
__global__ void k_output(const float* __restrict__ dplane, const float* __restrict__ Xn,
                         float* __restrict__ out) {
    int idx = blockIdx.x * blockDim.x + threadIdx.x;
    if (idx >= NC) return;
    float dX[9], D2[9];
    build33(dplane, idx, dX);
#pragma unroll
    for (int i = 0; i < 3; ++i)
#pragma unroll
        for (int j = 0; j < 3; ++j) {
            float t = 0.0f;
#pragma unroll
            for (int k = 0; k < 3; ++k) t += dX[i * 3 + k] * dX[k * 3 + j];
            D2[i * 3 + j] = t;
        }
    const float* xn = Xn + (size_t)idx * 9;
    float* o = out + (size_t)idx * 9;
#pragma unroll
    for (int i = 0; i < 9; ++i) o[i] = xn[i] + dX[i] + D2[i];
}

// ---------------- host launcher ----------------
extern "C" void kernel_launch(void* const* d_in, const int* in_sizes, int n_in,
                              void* d_out, int out_size, void* d_ws, size_t ws_size,
                              hipStream_t stream) {
    const float* X       = (const float*)d_in[0];
    const float* charges = (const float*)d_in[1];
    const float* ew      = (const float*)d_in[2];
    const float* ea      = (const float*)d_in[3];
    const float* W1      = (const float*)d_in[4];
    const float* b1      = (const float*)d_in[5];
    const float* W2      = (const float*)d_in[6];
    const float* b2      = (const float*)d_in[7];
    const float* W3      = (const float*)d_in[8];
    const float* b3      = (const float*)d_in[9];
    const float* Wt      = (const float*)d_in[10];
    const int*   eidx    = (const int*)d_in[11];

    float* ws     = (float*)d_ws;
    float* Xn     = ws;                              // NC*9 floats
    float* planeA = Xn + (size_t)NC * 9;             // 10*NC (pre-mix, later: scaled)
    float* planeB = planeA + (size_t)10 * NC;        // 10*NC (mixed I/A/S -> Y)
    float* planeC = planeB + (size_t)10 * NC;        // 10*NC (msg accum, later: dI/dA/dS)

    int eb = (NC + 255) / 256;                       // element-wise grid: 5000 blocks

    // 1) normalize + decompose X -> planeA, Xn
    k_decompose<<<eb, 256, 0, stream>>>(X, Xn, planeA);
    // 2) first channel mixing (Wt[0..2]) -> planeB
    k_mix<<<dim3(N_MBLK, 10), 256, 0, stream>>>(planeA, Wt, planeB, 0, 1, 2);
    // 3) zero message accumulators
    k_zero<<<(10 * NC + 255) / 256, 256, 0, stream>>>(planeC, 10 * NC);
    // 4) fused edge MLP (WMMA) + message scatter -> planeC
    k_edge<<<N_EDGES / ET, 256, 0, stream>>>(charges, ew, ea, W1, b1, W2, b2, W3, b3,
                                             eidx, planeB, planeC);
    // 5) T = msg@Y + Y@msg, decompose, scale by 1/(|T|^2+1) -> planeA
    k_recombine<<<eb, 256, 0, stream>>>(planeC, planeB, planeA);
    // 6) second channel mixing (Wt[3..5]) -> planeC
    k_mix<<<dim3(N_MBLK, 10), 256, 0, stream>>>(planeA, Wt, planeC, 3, 4, 5);
    // 7) out = Xn + dX + dX@dX
    k_output<<<eb, 256, 0, stream>>>(planeC, Xn, (float*)d_out);
}